// MiniBatchDiscrimination_1580547970506
// MI455X (gfx1250) — compile-verified
//
#include <hip/hip_runtime.h>
#include <hip/hip_bf16.h>

typedef __attribute__((ext_vector_type(2))) float v2f;
typedef __attribute__((ext_vector_type(8))) float v8f;

#define B_SZ       512
#define IN_F       512
#define OUT_F      64
#define K_DIMS     16
#define NCOLS      (OUT_F * K_DIMS)    // 1024
#define OUT_STRIDE (IN_F + OUT_F)      // 576

// ---------------------------------------------------------------------------
// Phase 1: M[512][1024] = X[512][512] @ W[512][1024], f32, via
// V_WMMA_F32_16X16X4_F32. One wave per 16x16 output tile; 2048 tiles total.
// A-frag (16x4 f32, 2 VGPR): lane<16 -> {K=k0,k0+1}, lane>=16 -> {K=k0+2,k0+3},
// row M = lane&15.  B-frag (4x16): same K split, col N = lane&15.
// C/D (16x16 f32, 8 VGPR): vgpr r holds M = r + 8*(lane>>4), N = lane&15.
// ---------------------------------------------------------------------------
__global__ __launch_bounds__(256)
void gemm_wmma_f32(const float* __restrict__ X, const float* __restrict__ W,
                   float* __restrict__ M) {
  const int tid   = threadIdx.x;
  const int lane  = tid & 31;
  const int wave  = tid >> 5;
  const int tile  = blockIdx.x * 8 + wave;   // 0..2047
  const int mt    = tile >> 6;               // 0..31 (row tile)
  const int nt    = tile & 63;               // 0..63 (col tile)
  const int r15   = lane & 15;
  const int khalf = lane >> 4;               // 0 or 1
  const int mrow  = mt * 16 + r15;
  const int ncol  = nt * 16 + r15;

  const float* __restrict__ xrow = X + mrow * IN_F;

  v8f acc0 = {}, acc1 = {}, acc2 = {}, acc3 = {};
  for (int k = 0; k < IN_F; k += 16) {
    const int ka = k + 2 * khalf;

    v2f a0 = *(const v2f*)(xrow + ka);
    v2f a1 = *(const v2f*)(xrow + ka + 4);
    v2f a2 = *(const v2f*)(xrow + ka + 8);
    v2f a3 = *(const v2f*)(xrow + ka + 12);

    v2f b0, b1, b2, b3;
    b0.x = W[(ka     ) * NCOLS + ncol];  b0.y = W[(ka +  1) * NCOLS + ncol];
    b1.x = W[(ka +  4) * NCOLS + ncol];  b1.y = W[(ka +  5) * NCOLS + ncol];
    b2.x = W[(ka +  8) * NCOLS + ncol];  b2.y = W[(ka +  9) * NCOLS + ncol];
    b3.x = W[(ka + 12) * NCOLS + ncol];  b3.y = W[(ka + 13) * NCOLS + ncol];

    acc0 = __builtin_amdgcn_wmma_f32_16x16x4_f32(false, a0, false, b0,
                                                 (short)0, acc0, false, false);
    acc1 = __builtin_amdgcn_wmma_f32_16x16x4_f32(false, a1, false, b1,
                                                 (short)0, acc1, false, false);
    acc2 = __builtin_amdgcn_wmma_f32_16x16x4_f32(false, a2, false, b2,
                                                 (short)0, acc2, false, false);
    acc3 = __builtin_amdgcn_wmma_f32_16x16x4_f32(false, a3, false, b3,
                                                 (short)0, acc3, false, false);
  }

  v8f acc = (acc0 + acc1) + (acc2 + acc3);
  float* mout = M + (mt * 16 + khalf * 8) * NCOLS + ncol;
  #pragma unroll
  for (int r = 0; r < 8; ++r) mout[r * NCOLS] = acc[r];
}

// ---------------------------------------------------------------------------
// Phase 2: o_b[q,o] = sum_p exp(-sum_k |m[p,o,k]-m[q,o,k]|) - 1, and copy x.
// 32 blocks x 256 threads; block handles q in [16*qt, 16*qt+16), all 64 o.
// Thread (o = tid&63, qg = tid>>6) accumulates 4 q's: q = 16*qt + qg + 4*i.
// Stages 8 p-rows (8*64*16 f32, padded) in LDS per barrier round.
// ---------------------------------------------------------------------------
#define P_TILE       8
#define LDS_O_STRIDE (P_TILE * K_DIMS + 4)   // 132 floats: 16B-aligned, banks split

__global__ __launch_bounds__(256)
void minibatch_l1exp(const float* __restrict__ X, const float* __restrict__ M,
                     float* __restrict__ OUT) {
  __shared__ float lds[OUT_F * LDS_O_STRIDE];   // 64*132*4 = 33792 bytes

  const int tid = threadIdx.x;
  const int qt  = blockIdx.x;     // 0..31
  const int o   = tid & 63;
  const int qg  = tid >> 6;       // 0..3

  // ---- copy x rows of this q-tile into OUT[:, 0:512] (float4) ----
  {
    const float4* xs = (const float4*)X;
    #pragma unroll
    for (int i = 0; i < 8; ++i) {
      const int idx = tid + i * 256;          // 0..2047 = 16 rows * 128 f4
      const int r = idx >> 7, c = idx & 127;
      *(float4*)(OUT + (qt * 16 + r) * OUT_STRIDE + c * 4) =
          xs[(qt * 16 + r) * (IN_F / 4) + c];
    }
  }

  // ---- load the 4 query fragments m[q,o,0..15] into registers ----
  float mq[4][K_DIMS];
  #pragma unroll
  for (int i = 0; i < 4; ++i) {
    const int q = qt * 16 + qg + i * 4;
    const float* src = M + (q * OUT_F + o) * K_DIMS;
    #pragma unroll
    for (int k = 0; k < K_DIMS; ++k) mq[i][k] = src[k];
  }

  float acc[4] = {0.f, 0.f, 0.f, 0.f};

  for (int pt = 0; pt < B_SZ / P_TILE; ++pt) {
    __syncthreads();
    // cooperative stage of 8 p-rows: 2048 float4, 8 per thread
    #pragma unroll
    for (int i = 0; i < 8; ++i) {
      const int f4  = tid + i * 256;    // 0..2047
      const int pl  = f4 >> 8;          // 0..7
      const int rem = f4 & 255;
      const int oo  = rem >> 2;         // 0..63
      const int kq  = rem & 3;          // float4 within k-dim
      float4 v = *(const float4*)(M + ((pt * P_TILE + pl) * OUT_F + oo) * K_DIMS + kq * 4);
      *(float4*)(&lds[oo * LDS_O_STRIDE + pl * K_DIMS + kq * 4]) = v;
    }
    __syncthreads();

    #pragma unroll
    for (int pl = 0; pl < P_TILE; ++pl) {
      const float* mp = &lds[o * LDS_O_STRIDE + pl * K_DIMS];
      float p0[K_DIMS];
      #pragma unroll
      for (int k = 0; k < K_DIMS; ++k) p0[k] = mp[k];   // 4x ds_read_b128
      #pragma unroll
      for (int i = 0; i < 4; ++i) {
        float n = 0.f;
        #pragma unroll
        for (int k = 0; k < K_DIMS; ++k) n += fabsf(p0[k] - mq[i][k]);
        acc[i] += __expf(-n);           // v_exp_f32
      }
    }
  }

  // self-term (p==q) contributed exp(0)=1; reference subtracts 1.
  #pragma unroll
  for (int i = 0; i < 4; ++i) {
    const int q = qt * 16 + qg + i * 4;
    OUT[q * OUT_STRIDE + IN_F + o] = acc[i] - 1.0f;
  }
}

// ---------------------------------------------------------------------------
extern "C" void kernel_launch(void* const* d_in, const int* in_sizes, int n_in,
                              void* d_out, int out_size, void* d_ws, size_t ws_size,
                              hipStream_t stream) {
  const float* x = (const float*)d_in[0];           // [512,512]
  const float* T = (const float*)d_in[1];           // [512,64,16] == [512,1024]
  float* out     = (float*)d_out;                   // [512,576]
  float* M       = (float*)d_ws;                    // needs 512*1024*4 = 2 MB

  gemm_wmma_f32<<<256, 256, 0, stream>>>(x, T, M);
  minibatch_l1exp<<<32, 256, 0, stream>>>(x, M, out);
}